// VecDGCNNAtten_86792699118038
// MI455X (gfx1250) — compile-verified
//
#include <hip/hip_runtime.h>
#include <math.h>

#define EPS        1e-6f
#define K_NN       20
#define NEG_SLOPE  0.2f

typedef __attribute__((ext_vector_type(2))) float v2f;
typedef __attribute__((ext_vector_type(8))) float v8f;

static constexpr int Bc = 8;
static constexpr int Nc = 2048;
static constexpr int Cc = 64;
static constexpr int NH = 4;      // 64 / ATTEN_C(16)
static constexpr int QT = Nc / 16;

// ---------------------------------------------------------------------------
// Kernel 1: squared norms of the kNN point cloud y
// ---------------------------------------------------------------------------
__global__ __launch_bounds__(256) void xx_kernel(const float* __restrict__ y,
                                                 float* __restrict__ xx) {
    int i = blockIdx.x * blockDim.x + threadIdx.x;      // b*N + n
    if (i >= Bc * Nc) return;
    int b = i / Nc, n = i % Nc;
    const float* yb = y + (size_t)b * 3 * Nc;
    float a = yb[0 * Nc + n], c = yb[1 * Nc + n], d = yb[2 * Nc + n];
    xx[i] = a * a + c * c + d * d;
}

// ---------------------------------------------------------------------------
// Kernel 2: kNN via V_WMMA_F32_16X16X4_F32 pairwise Gram tiles + top-20
// One wave (32 lanes) per 16 query rows.
// A: 16x4 f32 (lanes 0-15: K=0,1 ; lanes 16-31: K=2,3)
// B: 4x16 f32 (same split on K, lane%16 = column)
// D: 16x16 f32 (VGPR j: M=j (+8 for upper half-wave), N=lane%16)
// Top-20 list kept in registers, branchless select-chain insertion.
// ---------------------------------------------------------------------------
__global__ __launch_bounds__(32) void knn_kernel(const float* __restrict__ y,
                                                 const float* __restrict__ xx,
                                                 int* __restrict__ idx_out) {
    __shared__ float tile[16][17];

    int wg    = blockIdx.x;
    int b     = wg / QT;
    int qbase = (wg % QT) * 16;
    int l     = threadIdx.x;
    int half  = l >> 4;
    int m16   = l & 15;

    const float* yb  = y  + (size_t)b * 3 * Nc;
    const float* xxb = xx + (size_t)b * Nc;

    v2f afrag;
    if (half == 0) {
        afrag[0] = yb[0 * Nc + qbase + m16];
        afrag[1] = yb[1 * Nc + qbase + m16];
    } else {
        afrag[0] = yb[2 * Nc + qbase + m16];
        afrag[1] = 0.0f;
    }

    // register-resident top-k (owned by lanes 0-15, one row each)
    float bv[K_NN];
    int   bi[K_NN];
    #pragma unroll
    for (int t = 0; t < K_NN; ++t) { bv[t] = -3.0e38f; bi[t] = 0; }

    float xq = (l < 16) ? xxb[qbase + l] : 0.0f;

    for (int ct = 0; ct < QT; ++ct) {
        int cbase = ct * 16;
        v2f bfrag;
        if (half == 0) {
            bfrag[0] = yb[0 * Nc + cbase + m16];
            bfrag[1] = yb[1 * Nc + cbase + m16];
        } else {
            bfrag[0] = yb[2 * Nc + cbase + m16];
            bfrag[1] = 0.0f;
        }
        v8f acc = {};
        acc = __builtin_amdgcn_wmma_f32_16x16x4_f32(
                  false, afrag, false, bfrag, (short)0, acc, false, false);

        float xc = xxb[cbase + m16];                    // column bias (lane's column)
        #pragma unroll
        for (int j = 0; j < 8; ++j) {
            int m = j + half * 8;
            tile[m][m16] = 2.0f * acc[j] - xc;
        }
        __syncthreads();

        if (l < 16) {
            #pragma unroll
            for (int nn = 0; nn < 16; ++nn) {
                float v = tile[l][nn] - xq;             // neg squared distance
                if (v > bv[K_NN - 1]) {
                    // stable rank: elements >= v stay ahead (lower index first)
                    int pos = 0;
                    #pragma unroll
                    for (int t = 0; t < K_NN; ++t) pos += (bv[t] >= v) ? 1 : 0;
                    #pragma unroll
                    for (int t = K_NN - 1; t > 0; --t) {
                        bool sh = (t > pos);
                        bv[t] = sh ? bv[t - 1] : bv[t];
                        bi[t] = sh ? bi[t - 1] : bi[t];
                    }
                    int id = cbase + nn;
                    #pragma unroll
                    for (int t = 0; t < K_NN; ++t) {
                        bool put = (t == pos);
                        bv[t] = put ? v  : bv[t];
                        bi[t] = put ? id : bi[t];
                    }
                }
            }
        }
        __syncthreads();
    }

    if (l < 16) {
        int* op = idx_out + ((size_t)b * Nc + qbase + l) * K_NN;
        #pragma unroll
        for (int t = 0; t < K_NN; ++t) op[t] = bi[t];
    }
}

// ---------------------------------------------------------------------------
// Kernel 3: q_x = channel_equi_vec_normalize(vn_linear_leaky_relu(x, Wf1, Wd1))
// Block = 64 threads = channels, one block per (b,n).
// ---------------------------------------------------------------------------
__global__ __launch_bounds__(64) void qx_kernel(const float* __restrict__ x,
                                                const float* __restrict__ Wf1,
                                                const float* __restrict__ Wd1,
                                                float* __restrict__ qx) {
    __shared__ float red[Cc];
    int bn = blockIdx.x;
    int b = bn / Nc, n = bn % Nc;
    int o = threadIdx.x;

    const float* xb = x + (size_t)b * 3 * Nc;
    float x0 = xb[n], x1 = xb[Nc + n], x2 = xb[2 * Nc + n];
    float wf = Wf1[o], wd = Wd1[o];
    float s   = x0 * x0 + x1 * x1 + x2 * x2;
    float dot = wf * wd * s;
    float d2  = wd * wd * s;
    float r   = dot / (d2 + EPS);
    float cp  = (dot >= 0.0f) ? wf : (wf - r * wd);
    float coef = NEG_SLOPE * wf + (1.0f - NEG_SLOPE) * cp;
    float v0 = coef * x0, v1 = coef * x1, v2 = coef * x2;

    float n2 = v0 * v0 + v1 * v1 + v2 * v2;
    red[o] = n2;
    __syncthreads();
    float S = 0.0f;
    #pragma unroll
    for (int t = 0; t < Cc; ++t) S += red[t];

    float nrm = sqrtf(n2);
    float sc  = (1.0f / fmaxf(nrm, 1e-12f)) * (nrm / fmaxf(sqrtf(S), 1e-12f));
    float* q = qx + ((size_t)bn * Cc + o) * 3;
    q[0] = v0 * sc; q[1] = v1 * sc; q[2] = v2 * sc;
}

// ---------------------------------------------------------------------------
// Kernel 4: edge features -> VN leaky relu -> normalized qk -> head softmax
//           -> attention-weighted sum.  Block = 64 threads per (b,n).
// ---------------------------------------------------------------------------
__global__ __launch_bounds__(64) void atten_kernel(const float* __restrict__ y,
                                                   const float* __restrict__ Wf2,
                                                   const float* __restrict__ Wd2,
                                                   const int* __restrict__ knn,
                                                   const float* __restrict__ qx,
                                                   float* __restrict__ out) {
    __shared__ float yf[K_NN][Cc][3];
    __shared__ float qk[Cc][K_NN];
    __shared__ float hs[NH][K_NN];
    __shared__ float at[NH][K_NN];
    __shared__ float red[Cc];

    int bn = blockIdx.x;
    int b = bn / Nc, n = bn % Nc;
    int o = threadIdx.x;
    int h = o >> 4;

    const float* yb = y + (size_t)b * 3 * Nc;
    float c0 = yb[n], c1 = yb[Nc + n], c2 = yb[2 * Nc + n];
    float wf0 = Wf2[o * 2 + 0], wf1 = Wf2[o * 2 + 1];
    float wd0 = Wd2[o * 2 + 0], wd1 = Wd2[o * 2 + 1];
    const float* q = qx + ((size_t)bn * Cc + o) * 3;
    float q0 = q[0], q1 = q[1], q2 = q[2];

    for (int t = o; t < NH * K_NN; t += Cc) ((float*)hs)[t] = 0.0f;
    __syncthreads();

    for (int k = 0; k < K_NN; ++k) {
        int j = knn[(size_t)bn * K_NN + k];
        float e0 = yb[j] - c0, e1 = yb[Nc + j] - c1, e2 = yb[2 * Nc + j] - c2;
        float p0 = wf0 * e0 + wf1 * c0;
        float p1 = wf0 * e1 + wf1 * c1;
        float p2 = wf0 * e2 + wf1 * c2;
        float d0 = wd0 * e0 + wd1 * c0;
        float d1 = wd0 * e1 + wd1 * c1;
        float d2v = wd0 * e2 + wd1 * c2;
        float dot = p0 * d0 + p1 * d1 + p2 * d2v;
        float dd  = d0 * d0 + d1 * d1 + d2v * d2v;
        float r   = dot / (dd + EPS);
        float f0, f1, f2;
        if (dot >= 0.0f) { f0 = p0; f1 = p1; f2 = p2; }
        else { f0 = p0 - r * d0; f1 = p1 - r * d1; f2 = p2 - r * d2v; }
        f0 = NEG_SLOPE * p0 + (1.0f - NEG_SLOPE) * f0;
        f1 = NEG_SLOPE * p1 + (1.0f - NEG_SLOPE) * f1;
        f2 = NEG_SLOPE * p2 + (1.0f - NEG_SLOPE) * f2;
        yf[k][o][0] = f0; yf[k][o][1] = f1; yf[k][o][2] = f2;

        float n2 = f0 * f0 + f1 * f1 + f2 * f2;
        red[o] = n2;
        __syncthreads();
        float S = 0.0f;
        #pragma unroll
        for (int t = 0; t < Cc; ++t) S += red[t];
        float nrm = sqrtf(n2);
        float sc  = (1.0f / fmaxf(nrm, 1e-12f)) * (nrm / fmaxf(sqrtf(S), 1e-12f));
        qk[o][k] = (f0 * q0 + f1 * q1 + f2 * q2) * sc;
        __syncthreads();
    }

    // head sums via LDS float atomics (ds_add_f32)
    for (int k = 0; k < K_NN; ++k) atomicAdd(&hs[h][k], qk[o][k]);
    __syncthreads();

    if (o < NH) {
        const float inv = 0.14433756729740643f;   // 1/sqrt(3*16)
        float mx = -3.0e38f;
        #pragma unroll
        for (int k = 0; k < K_NN; ++k) mx = fmaxf(mx, hs[o][k] * inv);
        float se = 0.0f;
        #pragma unroll
        for (int k = 0; k < K_NN; ++k) {
            float e = __expf(hs[o][k] * inv - mx);
            at[o][k] = e; se += e;
        }
        float rs = 1.0f / se;
        #pragma unroll
        for (int k = 0; k < K_NN; ++k) at[o][k] *= rs;
    }
    __syncthreads();

    float o0 = 0.0f, o1 = 0.0f, o2 = 0.0f;
    #pragma unroll
    for (int k = 0; k < K_NN; ++k) {
        float a = at[h][k];
        o0 += a * yf[k][o][0];
        o1 += a * yf[k][o][1];
        o2 += a * yf[k][o][2];
    }
    float* op = out + (((size_t)b * Cc + o) * 3) * Nc + n;
    op[0] = o0; op[Nc] = o1; op[2 * Nc] = o2;
}

// ---------------------------------------------------------------------------
extern "C" void kernel_launch(void* const* d_in, const int* in_sizes, int n_in,
                              void* d_out, int out_size, void* d_ws, size_t ws_size,
                              hipStream_t stream) {
    const float* x   = (const float*)d_in[0];
    const float* y   = (const float*)d_in[1];
    const float* Wf1 = (const float*)d_in[2];
    const float* Wd1 = (const float*)d_in[3];
    const float* Wf2 = (const float*)d_in[4];
    const float* Wd2 = (const float*)d_in[5];
    float* out = (float*)d_out;

    char* ws = (char*)d_ws;
    float* xx  = (float*)ws;                                     // B*N floats
    int*   knn = (int*)(ws + (size_t)Bc * Nc * 4);               // B*N*K ints
    float* qx  = (float*)(ws + (size_t)Bc * Nc * 4
                             + (size_t)Bc * Nc * K_NN * 4);      // B*N*C*3 floats

    xx_kernel<<<(Bc * Nc + 255) / 256, 256, 0, stream>>>(y, xx);
    knn_kernel<<<Bc * QT, 32, 0, stream>>>(y, xx, knn);
    qx_kernel<<<Bc * Nc, 64, 0, stream>>>(x, Wf1, Wd1, qx);
    atten_kernel<<<Bc * Nc, 64, 0, stream>>>(y, Wf2, Wd2, knn, qx, out);
}